// HeteroLinkPredictor_49280454754830
// MI455X (gfx1250) — compile-verified
//
#include <hip/hip_runtime.h>
#include <hip/hip_bf16.h>

typedef float v2f __attribute__((ext_vector_type(2)));
typedef float v8f __attribute__((ext_vector_type(8)));

#define HID 64

// ---------------------------------------------------------------------------
// WMMA helpers: V_WMMA_F32_16X16X4_F32  (D = A(16x4) * B(4x16) + C(16x16))
// A layout (32-bit, 16x4): lanes 0-15 -> M=lane, VGPR0=K0,VGPR1=K1;
//                          lanes16-31 -> M=lane-16, VGPR0=K2,VGPR1=K3.
// B layout (4x16): lanes 0-15 -> N=lane, VGPR0=K0,VGPR1=K1; lanes16-31 -> K2,K3.
// C/D (16x16): VGPR v: lanes0-15 M=v, lanes16-31 M=v+8; N=lane&15.
// ---------------------------------------------------------------------------
__device__ __forceinline__ v8f wmma4(v2f a, v2f b, v8f c) {
  return __builtin_amdgcn_wmma_f32_16x16x4_f32(false, a, false, b, (short)0, c,
                                               false, false);
}

// A fragment from row-major [rows, ld] at (r0, k0)
__device__ __forceinline__ v2f load_a(const float* __restrict__ src, int ld,
                                      int r0, int k0, int lane) {
  int row = lane & 15;
  int koff = (lane >> 4) << 1;
  const float* p = src + (size_t)(r0 + row) * ld + k0 + koff;
  v2f a; a.x = p[0]; a.y = p[1];
  return a;
}

// B fragment = W^T tile: B[k,n] = W[n,k], W row-major [Nout, K(ld)]
__device__ __forceinline__ v2f load_bT(const float* __restrict__ W, int ld,
                                       int n0, int k0, int lane) {
  int n = lane & 15;
  int koff = (lane >> 4) << 1;
  const float* p = W + (size_t)(n0 + n) * ld + k0 + koff;
  v2f b; b.x = p[0]; b.y = p[1];
  return b;
}

// ---------------------------------------------------------------------------
// Utility kernels
// ---------------------------------------------------------------------------
__global__ void zero_kernel(float* __restrict__ p, int n) {
  int stride = gridDim.x * blockDim.x;
  for (int i = blockIdx.x * blockDim.x + threadIdx.x; i < n; i += stride)
    p[i] = 0.0f;
}

// x_user[i,:] = user_emb[user_node_id[i],:]   (float4 granularity)
__global__ void gather_rows_kernel(float* __restrict__ dst,
                                   const float* __restrict__ src,
                                   const int* __restrict__ idx, int total16) {
  int t = blockIdx.x * blockDim.x + threadIdx.x;
  if (t >= total16) return;
  int row = t >> 4;
  int c = (t & 15) << 2;
  const float4 v = *(const float4*)(src + (size_t)idx[row] * HID + c);
  *(float4*)(dst + (size_t)row * HID + c) = v;
}

// degree counts (same graph for both layers)
__global__ void count_kernel(float* __restrict__ cnt_u, float* __restrict__ cnt_m,
                             const int* __restrict__ src_u,
                             const int* __restrict__ dst_m, int E) {
  int e = blockIdx.x * blockDim.x + threadIdx.x;
  if (e >= E) return;
  unsafeAtomicAdd(cnt_u + src_u[e], 1.0f);
  unsafeAtomicAdd(cnt_m + dst_m[e], 1.0f);
}

// agg[dst[e], :] += x_src[src[e], :]  ; thread = (edge, 4-float chunk)
__global__ void scatter_add_kernel(float* __restrict__ agg,
                                   const float* __restrict__ xsrc,
                                   const int* __restrict__ src_idx,
                                   const int* __restrict__ dst_idx,
                                   unsigned int total) {
  unsigned int t = blockIdx.x * blockDim.x + threadIdx.x;
  if (t >= total) return;
  int e = (int)(t >> 4);
  int c = ((int)t & 15) << 2;
  int s = src_idx[e];
  int d = dst_idx[e];
  const float4 v = *(const float4*)(xsrc + (size_t)s * HID + c);
  float* p = agg + (size_t)d * HID + c;
  unsafeAtomicAdd(p + 0, v.x);
  unsafeAtomicAdd(p + 1, v.y);
  unsafeAtomicAdd(p + 2, v.z);
  unsafeAtomicAdd(p + 3, v.w);
}

// ---------------------------------------------------------------------------
// x_movie = movie_x @ lin_W^T + lin_b + movie_emb[movie_node_id]
// one wave per 16-movie strip; WMMA f32 16x16x4, K=128
// ---------------------------------------------------------------------------
__global__ void __launch_bounds__(256)
movie_init_kernel(float* __restrict__ x_movie, const float* __restrict__ movie_x,
                  const float* __restrict__ lin_W, const float* __restrict__ lin_b,
                  const float* __restrict__ movie_emb,
                  const int* __restrict__ movie_nid, int n_movies, int MF) {
  int lane = threadIdx.x & 31;
  int wave = (blockIdx.x * blockDim.x + threadIdx.x) >> 5;
  int waves = (gridDim.x * blockDim.x) >> 5;
  int nstrips = n_movies >> 4;
  for (int s = wave; s < nstrips; s += waves) {
    int r0 = s << 4;
    v8f c[4] = {};
    for (int k0 = 0; k0 < MF; k0 += 4) {
      v2f a = load_a(movie_x, MF, r0, k0, lane);
#pragma unroll
      for (int nt = 0; nt < 4; ++nt)
        c[nt] = wmma4(a, load_bT(lin_W, MF, nt << 4, k0, lane), c[nt]);
    }
    int col = lane & 15;
    int rbase = (lane >> 4) << 3;
#pragma unroll
    for (int nt = 0; nt < 4; ++nt) {
      int n = (nt << 4) + col;
      float bias = lin_b[n];
#pragma unroll
      for (int v = 0; v < 8; ++v) {
        int row = r0 + rbase + v;
        int mid = movie_nid[row];
        x_movie[(size_t)row * HID + n] =
            c[nt][v] + bias + movie_emb[(size_t)mid * HID + n];
      }
    }
  }
}

// ---------------------------------------------------------------------------
// out = (agg/max(cnt,1)) @ W_l^T + b_l + x_dst @ W_r^T  [+ relu]
// one wave per 16-node strip; A fragments (K=64) held in 32 VGPRs
// ---------------------------------------------------------------------------
__global__ void __launch_bounds__(256)
sage_finish_kernel(float* __restrict__ out, const float* __restrict__ agg,
                   const float* __restrict__ cnt, const float* __restrict__ xdst,
                   const float* __restrict__ W_l, const float* __restrict__ b_l,
                   const float* __restrict__ W_r, int n_dst, int do_relu) {
  int lane = threadIdx.x & 31;
  int wave = (blockIdx.x * blockDim.x + threadIdx.x) >> 5;
  int waves = (gridDim.x * blockDim.x) >> 5;
  int nstrips = n_dst >> 4;
  int row = lane & 15;
  int koff = (lane >> 4) << 1;
  for (int s = wave; s < nstrips; s += waves) {
    int r0 = s << 4;
    float inv = 1.0f / fmaxf(cnt[r0 + row], 1.0f);
    v2f Af[16];
    const float* arow = agg + (size_t)(r0 + row) * HID + koff;
#pragma unroll
    for (int k = 0; k < 16; ++k) {
      Af[k].x = arow[(k << 2) + 0] * inv;
      Af[k].y = arow[(k << 2) + 1] * inv;
    }
    v8f c[4] = {};
#pragma unroll
    for (int nt = 0; nt < 4; ++nt)
#pragma unroll
      for (int k = 0; k < 16; ++k)
        c[nt] = wmma4(Af[k], load_bT(W_l, HID, nt << 4, k << 2, lane), c[nt]);

    const float* xrow = xdst + (size_t)(r0 + row) * HID + koff;
#pragma unroll
    for (int k = 0; k < 16; ++k) {
      Af[k].x = xrow[(k << 2) + 0];
      Af[k].y = xrow[(k << 2) + 1];
    }
#pragma unroll
    for (int nt = 0; nt < 4; ++nt)
#pragma unroll
      for (int k = 0; k < 16; ++k)
        c[nt] = wmma4(Af[k], load_bT(W_r, HID, nt << 4, k << 2, lane), c[nt]);

    int col = lane & 15;
    int rbase = (lane >> 4) << 3;
#pragma unroll
    for (int nt = 0; nt < 4; ++nt) {
      float bias = b_l[(nt << 4) + col];
#pragma unroll
      for (int v = 0; v < 8; ++v) {
        float x = c[nt][v] + bias;
        if (do_relu) x = fmaxf(x, 0.0f);
        out[(size_t)(r0 + rbase + v) * HID + (nt << 4) + col] = x;
      }
    }
  }
}

// ---------------------------------------------------------------------------
// decoder: out[i] = dot(o_user[lu[i]], o_movie[lm[i]])
// ---------------------------------------------------------------------------
__global__ void decode_kernel(float* __restrict__ out,
                              const float* __restrict__ ou,
                              const float* __restrict__ om,
                              const int* __restrict__ lu,
                              const int* __restrict__ lm, int L) {
  int i = blockIdx.x * blockDim.x + threadIdx.x;
  if (i >= L) return;
  const float4* a = (const float4*)(ou + (size_t)lu[i] * HID);
  const float4* b = (const float4*)(om + (size_t)lm[i] * HID);
  float s = 0.0f;
#pragma unroll
  for (int k = 0; k < 16; ++k) {
    float4 x = a[k], y = b[k];
    s += x.x * y.x + x.y * y.y + x.z * y.z + x.w * y.w;
  }
  out[i] = s;
}

// ---------------------------------------------------------------------------
extern "C" void kernel_launch(void* const* d_in, const int* in_sizes, int n_in,
                              void* d_out, int out_size, void* d_ws, size_t ws_size,
                              hipStream_t stream) {
  const float* movie_x   = (const float*)d_in[0];
  const int*   user_nid  = (const int*)d_in[1];
  const int*   movie_nid = (const int*)d_in[2];
  const int*   e_src_u   = (const int*)d_in[3];
  const int*   e_dst_m   = (const int*)d_in[4];
  const int*   l_src_u   = (const int*)d_in[5];
  const int*   l_dst_m   = (const int*)d_in[6];
  const float* user_emb  = (const float*)d_in[7];
  const float* movie_emb = (const float*)d_in[8];
  const float* lin_W     = (const float*)d_in[9];
  const float* lin_b     = (const float*)d_in[10];
  const float* W_l_um1 = (const float*)d_in[11];
  const float* b_l_um1 = (const float*)d_in[12];
  const float* W_r_um1 = (const float*)d_in[13];
  const float* W_l_mu1 = (const float*)d_in[14];
  const float* b_l_mu1 = (const float*)d_in[15];
  const float* W_r_mu1 = (const float*)d_in[16];
  const float* W_l_um2 = (const float*)d_in[17];
  const float* b_l_um2 = (const float*)d_in[18];
  const float* W_r_um2 = (const float*)d_in[19];
  const float* W_l_mu2 = (const float*)d_in[20];
  const float* b_l_mu2 = (const float*)d_in[21];
  const float* W_r_mu2 = (const float*)d_in[22];

  const int U  = in_sizes[7] / HID;   // 100000
  const int M  = in_sizes[8] / HID;   // 50000
  const int E  = in_sizes[3];         // 2000000
  const int L  = in_sizes[5];         // 500000
  const int MF = in_sizes[0] / M;     // 128

  // workspace layout
  float* ws      = (float*)d_ws;
  float* x_user  = ws;
  float* x_movie = x_user  + (size_t)U * HID;
  float* h_user  = x_movie + (size_t)M * HID;
  float* h_movie = h_user  + (size_t)U * HID;
  float* o_user  = h_movie + (size_t)M * HID;
  float* o_movie = o_user  + (size_t)U * HID;
  float* agg_u   = o_movie + (size_t)M * HID;
  float* agg_m   = agg_u   + (size_t)U * HID;
  float* cnt_u   = agg_m   + (size_t)M * HID;
  float* cnt_m   = cnt_u   + (size_t)U;

  const int TB = 256;
  const unsigned int scat_total = (unsigned int)E * 16u;
  const int scat_blocks  = (int)((scat_total + TB - 1) / TB);
  const int strips_m = M >> 4, strips_u = U >> 4;
  const int gemm_blocks_m = (strips_m + 7) / 8;   // 8 waves / block
  const int gemm_blocks_u = (strips_u + 7) / 8;

  // degree counts (shared by both layers)
  zero_kernel<<<1024, TB, 0, stream>>>(cnt_u, U);
  zero_kernel<<<1024, TB, 0, stream>>>(cnt_m, M);
  count_kernel<<<(E + TB - 1) / TB, TB, 0, stream>>>(cnt_u, cnt_m, e_src_u, e_dst_m, E);

  // feature init
  gather_rows_kernel<<<(U * 16 + TB - 1) / TB, TB, 0, stream>>>(x_user, user_emb, user_nid, U * 16);
  movie_init_kernel<<<gemm_blocks_m, TB, 0, stream>>>(x_movie, movie_x, lin_W, lin_b,
                                                      movie_emb, movie_nid, M, MF);

  // ---- layer 1: user -> movie ----
  zero_kernel<<<4096, TB, 0, stream>>>(agg_m, M * HID);
  scatter_add_kernel<<<scat_blocks, TB, 0, stream>>>(agg_m, x_user, e_src_u, e_dst_m, scat_total);
  sage_finish_kernel<<<gemm_blocks_m, TB, 0, stream>>>(h_movie, agg_m, cnt_m, x_movie,
                                                       W_l_um1, b_l_um1, W_r_um1, M, 1);
  // ---- layer 1: movie -> user ----
  zero_kernel<<<4096, TB, 0, stream>>>(agg_u, U * HID);
  scatter_add_kernel<<<scat_blocks, TB, 0, stream>>>(agg_u, x_movie, e_dst_m, e_src_u, scat_total);
  sage_finish_kernel<<<gemm_blocks_u, TB, 0, stream>>>(h_user, agg_u, cnt_u, x_user,
                                                       W_l_mu1, b_l_mu1, W_r_mu1, U, 1);

  // ---- layer 2: user -> movie ----
  zero_kernel<<<4096, TB, 0, stream>>>(agg_m, M * HID);
  scatter_add_kernel<<<scat_blocks, TB, 0, stream>>>(agg_m, h_user, e_src_u, e_dst_m, scat_total);
  sage_finish_kernel<<<gemm_blocks_m, TB, 0, stream>>>(o_movie, agg_m, cnt_m, h_movie,
                                                       W_l_um2, b_l_um2, W_r_um2, M, 0);
  // ---- layer 2: movie -> user ----
  zero_kernel<<<4096, TB, 0, stream>>>(agg_u, U * HID);
  scatter_add_kernel<<<scat_blocks, TB, 0, stream>>>(agg_u, h_movie, e_dst_m, e_src_u, scat_total);
  sage_finish_kernel<<<gemm_blocks_u, TB, 0, stream>>>(o_user, agg_u, cnt_u, h_user,
                                                       W_l_mu2, b_l_mu2, W_r_mu2, U, 0);

  // decoder
  decode_kernel<<<(L + TB - 1) / TB, TB, 0, stream>>>((float*)d_out, o_user, o_movie,
                                                      l_src_u, l_dst_m, L);
}